// transformationMatSingle_7653631721981
// MI455X (gfx1250) — compile-verified
//
#include <hip/hip_runtime.h>

typedef float v2f __attribute__((ext_vector_type(2)));
typedef float v4f __attribute__((ext_vector_type(4)));
typedef float v8f __attribute__((ext_vector_type(8)));

#define N_IN   256
#define BATCH  4096

// ---------------------------------------------------------------------------
// Kernel 1: build the 256 4x4 transform matrices (Rodrigues) into d_ws.
// wm layout: wm[n*16 + r*4 + c], row-major 4x4 per n.
// ---------------------------------------------------------------------------
__global__ __launch_bounds__(256) void build_wm_kernel(
    const float* __restrict__ aW, const float* __restrict__ uW,
    const float* __restrict__ tW, float* __restrict__ wm) {
  const int n = threadIdx.x;            // one thread per n (256 threads)
  const float a  = aW[n];
  const float sx = uW[n * 3 + 0];
  const float sy = uW[n * 3 + 1];
  const float sz = uW[n * 3 + 2];
  // sigmoid
  float ux = 1.0f / (1.0f + __expf(-sx));
  float uy = 1.0f / (1.0f + __expf(-sy));
  float uz = 1.0f / (1.0f + __expf(-sz));
  // normalize
  const float inv = rsqrtf(ux * ux + uy * uy + uz * uz);
  ux *= inv; uy *= inv; uz *= inv;
  const float s = __sinf(a);
  const float c = __cosf(a);
  // R = K*sin + (I - uuT)*cos + uuT
  const float R00 = (1.0f - ux * ux) * c + ux * ux;
  const float R01 = -uz * s - ux * uy * c + ux * uy;
  const float R02 =  uy * s - ux * uz * c + ux * uz;
  const float R10 =  uz * s - uy * ux * c + uy * ux;
  const float R11 = (1.0f - uy * uy) * c + uy * uy;
  const float R12 = -ux * s - uy * uz * c + uy * uz;
  const float R20 = -uy * s - uz * ux * c + uz * ux;
  const float R21 =  ux * s - uz * uy * c + uz * uy;
  const float R22 = (1.0f - uz * uz) * c + uz * uz;

  float* W = wm + n * 16;
  W[0]  = R00; W[1]  = R01; W[2]  = R02; W[3]  = tW[n * 3 + 0];
  W[4]  = R10; W[5]  = R11; W[6]  = R12; W[7]  = tW[n * 3 + 1];
  W[8]  = R20; W[9]  = R21; W[10] = R22; W[11] = tW[n * 3 + 2];
  W[12] = 0.0f; W[13] = 0.0f; W[14] = 0.0f; W[15] = 1.0f;
}

// ---------------------------------------------------------------------------
// Kernel 2: streaming I @ WM using V_WMMA_F32_16X16X4_F32.
//
// Per wave: fixed group g of 4 consecutive n's; 64 consecutive batches,
// manually unrolled 4x so four independent 256B loads are in flight before
// the first dependent WMMA (partial s_wait_loadcnt instead of full stalls).
//
//  A (16x4): A[i,k] = WM[g*4 + (i>>2)][k][i&3]      (loaded once, reused)
//  B (4x16): B[k,j] = I[group tile float j*4 + k]   (one b64/lane, 256B/wave)
//  D[i,j]   = (I_row_j @ WM[j>>2])[i&3] on block diagonal (i>>2 == j>>2);
//  C/D layout puts each output row contiguously in 4 VGPRs of one lane ->
//  predicated global_store_b128 writes each 256B out tile contiguously.
//
// VGPR layouts assumed per ISA 7.12.2 (wave32):
//  A 16x4 f32:  lane t row M=t&15; V0=A[M,2*(t>>4)],   V1=A[M,2*(t>>4)+1]
//  B 4x16 f32:  lane t col N=t&15; V0=B[2*(t>>4),N],   V1=B[2*(t>>4)+1,N]
//  D 16x16 f32: vgpr v, lanes0-15: M=v,N=lane; lanes16-31: M=v+8,N=lane-16
// ---------------------------------------------------------------------------
__global__ __launch_bounds__(256) void apply_wm_kernel(
    const float* __restrict__ I, const float* __restrict__ wm,
    float* __restrict__ out) {
  const int lane  = threadIdx.x & 31;
  const int wave  = (int)(blockIdx.x * 8u + (threadIdx.x >> 5)); // 0..4095
  const int g     = wave & 63;          // n-group: n = 4g .. 4g+3
  const int b0    = (wave >> 6) * 64;   // 64 consecutive batch items

  // ---- A operand: WM columns, replicated per 4-row block (loaded once) ----
  const int m  = (lane & 15) >> 2;     // which matrix of the group
  const int cc = lane & 3;             // output column handled by this row
  const int k0 = (lane >> 4) * 2;      // K-pair for this half-wave
  const int nIdx = g * 4 + m;
  v2f A;
  A.x = wm[nIdx * 16 + (k0 + 0) * 4 + cc];   // WM[k0  ][cc]
  A.y = wm[nIdx * 16 + (k0 + 1) * 4 + cc];   // WM[k0+1][cc]

  // per-lane float offset inside a 256B (64-float) group tile
  const int laneOff = (lane & 15) * 4 + (lane >> 4) * 2;

  // which lanes own a contiguous output row, and where it lives in D
  const bool storeActive = (lane < 8) || (lane >= 24);
  const bool useHi       = (lane & 4) != 0;
  const int  jrow        = lane & 15;

  const size_t groupBase = (size_t)g * 4 * 16;     // floats, within one batch
  const size_t bStride   = (size_t)N_IN * 16;      // floats per batch item

  const float* src = I   + (size_t)b0 * bStride + groupBase + laneOff;
  float*       dst = out + (size_t)b0 * bStride + groupBase + (size_t)jrow * 4;

  for (int it = 0; it < 16; ++it) {
    // prefetch the next unroll step (64 KB ahead): global_prefetch_b8
    __builtin_prefetch(src + 4 * bStride, 0, 1);

    // four independent 256B wave-loads in flight before any WMMA
    v2f B0 = __builtin_nontemporal_load((const v2f*)(src + 0 * bStride));
    v2f B1 = __builtin_nontemporal_load((const v2f*)(src + 1 * bStride));
    v2f B2 = __builtin_nontemporal_load((const v2f*)(src + 2 * bStride));
    v2f B3 = __builtin_nontemporal_load((const v2f*)(src + 3 * bStride));

    v8f Z = {0.f, 0.f, 0.f, 0.f, 0.f, 0.f, 0.f, 0.f};
    v8f C0 = __builtin_amdgcn_wmma_f32_16x16x4_f32(false, A, false, B0,
                                                   (short)0, Z, false, false);
    v8f C1 = __builtin_amdgcn_wmma_f32_16x16x4_f32(false, A, false, B1,
                                                   (short)0, Z, false, false);
    v8f C2 = __builtin_amdgcn_wmma_f32_16x16x4_f32(false, A, false, B2,
                                                   (short)0, Z, false, false);
    v8f C3 = __builtin_amdgcn_wmma_f32_16x16x4_f32(false, A, false, B3,
                                                   (short)0, Z, false, false);

    if (storeActive) {
      v4f r0 = useHi ? __builtin_shufflevector(C0, C0, 4, 5, 6, 7)
                     : __builtin_shufflevector(C0, C0, 0, 1, 2, 3);
      v4f r1 = useHi ? __builtin_shufflevector(C1, C1, 4, 5, 6, 7)
                     : __builtin_shufflevector(C1, C1, 0, 1, 2, 3);
      v4f r2 = useHi ? __builtin_shufflevector(C2, C2, 4, 5, 6, 7)
                     : __builtin_shufflevector(C2, C2, 0, 1, 2, 3);
      v4f r3 = useHi ? __builtin_shufflevector(C3, C3, 4, 5, 6, 7)
                     : __builtin_shufflevector(C3, C3, 0, 1, 2, 3);
      __builtin_nontemporal_store(r0, (v4f*)(dst + 0 * bStride));
      __builtin_nontemporal_store(r1, (v4f*)(dst + 1 * bStride));
      __builtin_nontemporal_store(r2, (v4f*)(dst + 2 * bStride));
      __builtin_nontemporal_store(r3, (v4f*)(dst + 3 * bStride));
    }

    src += 4 * bStride;
    dst += 4 * bStride;
  }
}

// ---------------------------------------------------------------------------
// Launch: inputs are (I, aW, uW, tW) f32; output f32 (4096,256,4,4).
// d_ws holds the 256 precomputed 4x4 matrices (16 KB).
// ---------------------------------------------------------------------------
extern "C" void kernel_launch(void* const* d_in, const int* in_sizes, int n_in,
                              void* d_out, int out_size, void* d_ws, size_t ws_size,
                              hipStream_t stream) {
  const float* I  = (const float*)d_in[0];
  const float* aW = (const float*)d_in[1];
  const float* uW = (const float*)d_in[2];
  const float* tW = (const float*)d_in[3];
  float* out = (float*)d_out;
  float* wm  = (float*)d_ws;

  build_wm_kernel<<<1, 256, 0, stream>>>(aW, uW, tW, wm);
  // 4096 waves total: 512 blocks x 8 waves; each wave: one n-group, 64 batches
  apply_wm_kernel<<<512, 256, 0, stream>>>(I, wm, out);
}